// S6_35433480192349
// MI455X (gfx1250) — compile-verified
//
#include <hip/hip_runtime.h>
#include <math.h>

#define BT 8
#define LSEQ 8192
#define FDIM 128
#define NST 16
#define CHUNK 128
#define NCH (LSEQ / CHUNK)

typedef float v2f __attribute__((ext_vector_type(2)));
typedef float v8f __attribute__((ext_vector_type(8)));
typedef int   v4i __attribute__((ext_vector_type(4)));
typedef __attribute__((address_space(1))) v4i* gv4i_p;   // global int4*
typedef __attribute__((address_space(3))) v4i* lv4i_p;   // LDS int4*

#if __has_builtin(__builtin_amdgcn_global_load_async_to_lds_b128)
#define HAS_ASYNC_LDS 1
#else
#define HAS_ASYNC_LDS 0
#endif

// async global->LDS 16-byte copy (ASYNCcnt path); falls back to a plain copy
__device__ __forceinline__ void async_copy_b128(const float* g, float* l) {
#if HAS_ASYNC_LDS
  __builtin_amdgcn_global_load_async_to_lds_b128((gv4i_p)g, (lv4i_p)l, 0, 0);
#else
  *(float4*)l = *(const float4*)g;
#endif
}

__device__ __forceinline__ void wait_async_all() {
#if HAS_ASYNC_LDS
  asm volatile("s_wait_asynccnt 0x0" ::: "memory");
#endif
}

// allow the 11 async ops of the next stage to stay in flight
__device__ __forceinline__ void wait_async_prev_stage() {
#if HAS_ASYNC_LDS
  asm volatile("s_wait_asynccnt 0xb" ::: "memory");
#endif
}

__device__ __forceinline__ float lane_read(float v, int srcLane) {
  // wave32 broadcast via ds_bpermute (byte index)
  return __int_as_float(__builtin_amdgcn_ds_bpermute(srcLane << 2, __float_as_int(v)));
}

__device__ __forceinline__ float softplus_f(float z) {
  return (z > 20.f) ? z : log1pf(expf(z));
}

// -------------------------------------------------------------------------
// K1: input-dependent SSM params.
//   B = x@Wb+bb, C = x@Wc+bc (fp32 WMMA 16x16x4), delta = softplus(x@Wd+bd),
//   dA = exp(delta*A), dB = (dA-1)/A * B.  Outputs [BT*L, N] arrays.
// One 128-thread block = 4 waves = 4 independent 16-row tiles.
// -------------------------------------------------------------------------
__global__ __launch_bounds__(128) void k1_proj(
    const float* __restrict__ x, const float* __restrict__ Wb, const float* __restrict__ bb,
    const float* __restrict__ Wc, const float* __restrict__ bc, const float* __restrict__ Wd,
    const float* __restrict__ bd, const float* __restrict__ Avec,
    float* __restrict__ dA_ws, float* __restrict__ dB_ws, float* __restrict__ C_ws) {
  __shared__ float xs[4][16][132];   // pad 128->132 (528B rows, 16B aligned, conflict-free)
  __shared__ float swb[128][16];
  __shared__ float swc[128][16];
  const int tid = threadIdx.x;
  const int rowBase = blockIdx.x * 64;   // 64 flattened (bt*L+l) rows per block

  // stage x tile + weights via async global->LDS
  {
    const float* gx = x + (size_t)rowBase * FDIM;
    #pragma unroll
    for (int k = 0; k < 8; ++k) {
      const int i4 = k * 128 + tid;            // 1024 float4 chunks
      const int row = i4 >> 5, col = (i4 & 31) * 4;
      async_copy_b128(gx + (size_t)i4 * 4, &xs[0][0][0] + row * 132 + col);
    }
    #pragma unroll
    for (int k = 0; k < 4; ++k) {
      const int i4 = k * 128 + tid;            // 512 float4 chunks each
      async_copy_b128(Wb + (size_t)i4 * 4, &swb[0][0] + i4 * 4);
      async_copy_b128(Wc + (size_t)i4 * 4, &swc[0][0] + i4 * 4);
    }
  }
  wait_async_all();
  __syncthreads();

  const int w    = tid >> 5;   // wave -> tile
  const int lane = tid & 31;
  const int m    = lane & 15;  // A-row / B-col / D-col
  const int hi   = lane >> 4;
  const int k0   = hi * 2;     // K sub-offset for upper half-wave

  v8f accB = {}; v8f accC = {};
  #pragma unroll 4
  for (int kk = 0; kk < 32; ++kk) {
    const int kb = kk * 4 + k0;
    v2f a;  a.x  = xs[w][m][kb];  a.y  = xs[w][m][kb + 1];
    v2f bB; bB.x = swb[kb][m];    bB.y = swb[kb + 1][m];
    v2f bC; bC.x = swc[kb][m];    bC.y = swc[kb + 1][m];
    accB = __builtin_amdgcn_wmma_f32_16x16x4_f32(false, a, false, bB, (short)0, accB, false, false);
    accC = __builtin_amdgcn_wmma_f32_16x16x4_f32(false, a, false, bC, (short)0, accC, false, false);
  }

  // delta for row m: each half-wave sums 64 of the 128 K-terms, then xor-16 combine
  float z = 0.f;
  for (int k = hi * 64; k < hi * 64 + 64; ++k) z += xs[w][m][k] * Wd[k];
  z += lane_read(z, lane ^ 16);
  const float dlt = softplus_f(z + bd[0]);

  const float an  = Avec[m];     // n == lane&15 for D-matrix columns
  const float bbn = bb[m], bcn = bc[m];
  #pragma unroll
  for (int v = 0; v < 8; ++v) {
    const int m2 = v + hi * 8;               // D-matrix row (time index in tile)
    const float dv  = lane_read(dlt, m2);    // delta of row m2 lives in lane m2
    const float dAv = expf(dv * an);
    const float Bv  = accB[v] + bbn;
    const float Cv  = accC[v] + bcn;
    const float dBv = (dAv - 1.f) / an * Bv;
    const size_t idx = (size_t)(rowBase + w * 16 + m2) * NST + m;
    dA_ws[idx] = dAv; dB_ws[idx] = dBv; C_ws[idx] = Cv;
  }
}

// -------------------------------------------------------------------------
// K2a: per-chunk local scan (h0 = 0). 512 independent blocks (bt,chunk).
// Thread t owns feature f=t and its full 16-state vector in registers.
// Double-buffered async global->LDS staging: stage sub+1 overlaps compute
// of sub; each stage = 11 async b128 ops/thread -> s_wait_asynccnt 11.
// Also emits chunk carry product P[n] and cumPC[t,n] = C_t[n]*prod dA.
// -------------------------------------------------------------------------
__global__ __launch_bounds__(128) void k2a_scan(
    const float* __restrict__ x,
    const float* __restrict__ dA_ws, const float* __restrict__ dB_ws, const float* __restrict__ C_ws,
    float* __restrict__ out, float* __restrict__ cumPC_ws,
    float* __restrict__ S_ws, float* __restrict__ P_ws) {
  __shared__ float sx[2][32][128];
  __shared__ float sdA[2][32][16], sdB[2][32][16], sC[2][32][16];
  const int tid = threadIdx.x;           // == f
  const int bt  = blockIdx.x / NCH;
  const int ch  = blockIdx.x % NCH;
  const size_t row0 = (size_t)bt * LSEQ + (size_t)ch * CHUNK;

  auto issue_stage = [&](int buf, int sub) {
    const size_t rsub = row0 + sub * 32;
    const float* gx = x + rsub * FDIM;
    #pragma unroll
    for (int k = 0; k < 8; ++k) {                       // 4096 floats of x
      const int i4 = k * 128 + tid;
      async_copy_b128(gx + (size_t)i4 * 4, &sx[buf][0][0] + i4 * 4);
    }
    async_copy_b128(dA_ws + rsub * NST + tid * 4, &sdA[buf][0][0] + tid * 4);
    async_copy_b128(dB_ws + rsub * NST + tid * 4, &sdB[buf][0][0] + tid * 4);
    async_copy_b128(C_ws  + rsub * NST + tid * 4, &sC[buf][0][0]  + tid * 4);
  };

  float h[NST];
  #pragma unroll
  for (int n = 0; n < NST; ++n) h[n] = 0.f;
  float cumP = 1.f;

  issue_stage(0, 0);
  for (int sub = 0; sub < 4; ++sub) {
    const int buf = sub & 1;
    if (sub < 3) {
      issue_stage(buf ^ 1, sub + 1);
      wait_async_prev_stage();   // current stage's 11 ops retired; next 11 in flight
    } else {
      wait_async_all();
    }
    __syncthreads();

    const size_t rsub = row0 + sub * 32;
    for (int tt = 0; tt < 32; ++tt) {
      const float xf = sx[buf][tt][tid];
      float y = 0.f;
      #pragma unroll
      for (int n = 0; n < NST; ++n) {
        h[n] = sdA[buf][tt][n] * h[n] + sdB[buf][tt][n] * xf;
        y += h[n] * sC[buf][tt][n];
      }
      out[(rsub + tt) * FDIM + tid] = y;          // coalesced
      if (tid < NST) {
        cumP *= sdA[buf][tt][tid];
        cumPC_ws[(rsub + tt) * NST + tid] = sC[buf][tt][tid] * cumP;
      }
    }
    __syncthreads();   // protects buf^1 reuse by next issue_stage
  }
  const size_t sbase = (size_t)(bt * NCH + ch) * NST;
  #pragma unroll
  for (int n = 0; n < NST; ++n) S_ws[(sbase + n) * FDIM + tid] = h[n];  // [n][f] coalesced
  if (tid < NST) P_ws[sbase + tid] = cumP;
}

// -------------------------------------------------------------------------
// K2b: sequential cross-chunk combine per bt (64 tiny steps):
//   H_0 = 0 ; H_{c+1} = P_c (.)n H_c + S_c    (broadcast over f)
// Writes carry-in state H for every chunk, layout [bt][c][n][f].
// -------------------------------------------------------------------------
__global__ __launch_bounds__(128) void k2b_combine(
    const float* __restrict__ S_ws, const float* __restrict__ P_ws, float* __restrict__ H_ws) {
  __shared__ float sp[NST];
  const int tid = threadIdx.x;   // == f
  const int bt  = blockIdx.x;
  float H[NST];
  #pragma unroll
  for (int n = 0; n < NST; ++n) H[n] = 0.f;
  for (int c = 0; c < NCH; ++c) {
    const size_t base = (size_t)(bt * NCH + c) * NST;
    #pragma unroll
    for (int n = 0; n < NST; ++n) H_ws[(base + n) * FDIM + tid] = H[n];
    if (tid < NST) sp[tid] = P_ws[base + tid];
    __syncthreads();
    #pragma unroll
    for (int n = 0; n < NST; ++n) H[n] = sp[n] * H[n] + S_ws[(base + n) * FDIM + tid];
    __syncthreads();
  }
}

// -------------------------------------------------------------------------
// K3: carry correction as a per-chunk [128 x 16] @ [16 x 128] fp32 WMMA GEMM:
//   y[t,f] += sum_n cumPC[t,n] * H[f,n]
// 8 waves x 8 N-tiles x 4 wmma_f32_16x16x4_f32 each; operands async-staged.
// -------------------------------------------------------------------------
__global__ __launch_bounds__(256) void k3_corr(
    const float* __restrict__ cumPC_ws, const float* __restrict__ H_ws, float* __restrict__ out) {
  __shared__ float sG[128][20];   // pad 16->20: rows stay 16B aligned + conflict-free
  __shared__ float sH[16][128];   // H carry-in, [n][f]
  const int tid = threadIdx.x;
  const int bt  = blockIdx.x / NCH;
  const int ch  = blockIdx.x % NCH;
  const size_t row0  = (size_t)bt * LSEQ + (size_t)ch * CHUNK;
  const size_t hbase = (size_t)(bt * NCH + ch) * NST;

  #pragma unroll
  for (int k = 0; k < 2; ++k) {
    const int i4 = k * 256 + tid;               // 512 float4 chunks each
    const int row = i4 >> 2, col = (i4 & 3) * 4;
    async_copy_b128(cumPC_ws + row0 * NST + (size_t)i4 * 4, &sG[0][0] + row * 20 + col);
    async_copy_b128(H_ws + hbase * FDIM + (size_t)i4 * 4, &sH[0][0] + i4 * 4);
  }
  wait_async_all();
  __syncthreads();

  const int w    = tid >> 5;   // wave -> M-tile over t
  const int lane = tid & 31;
  const int m    = lane & 15;
  const int hi   = lane >> 4;
  const int k0   = hi * 2;
  const int t0   = w * 16;

  for (int f0 = 0; f0 < FDIM; f0 += 16) {
    v8f acc = {};
    #pragma unroll
    for (int kk = 0; kk < 4; ++kk) {
      const int kb = kk * 4 + k0;
      v2f a; a.x = sG[t0 + m][kb];   a.y = sG[t0 + m][kb + 1];
      v2f b; b.x = sH[kb][f0 + m];   b.y = sH[kb + 1][f0 + m];
      acc = __builtin_amdgcn_wmma_f32_16x16x4_f32(false, a, false, b, (short)0, acc, false, false);
    }
    #pragma unroll
    for (int v = 0; v < 8; ++v) {
      const int m2 = v + hi * 8;
      const size_t idx = (row0 + t0 + m2) * FDIM + f0 + m;
      out[idx] += acc[v];
    }
  }
}

extern "C" void kernel_launch(void* const* d_in, const int* in_sizes, int n_in,
                              void* d_out, int out_size, void* d_ws, size_t ws_size,
                              hipStream_t stream) {
  const float* x  = (const float*)d_in[0];
  const float* Wb = (const float*)d_in[1];
  const float* bb = (const float*)d_in[2];
  const float* Wc = (const float*)d_in[3];
  const float* bc = (const float*)d_in[4];
  const float* Wd = (const float*)d_in[5];
  const float* bd = (const float*)d_in[6];
  const float* Av = (const float*)d_in[7];
  float* out = (float*)d_out;
  float* ws  = (float*)d_ws;

  const size_t SZ1 = (size_t)BT * LSEQ * NST;        // 1,048,576 floats (4 MB)
  float* dA_ws    = ws;
  float* dB_ws    = ws + SZ1;
  float* C_ws     = ws + 2 * SZ1;
  float* cumPC_ws = ws + 3 * SZ1;
  float* S_ws     = ws + 4 * SZ1;                    // [bt][ch][n][f]
  float* P_ws     = ws + 5 * SZ1;                    // [bt][ch][n]  (32 KB)
  float* H_ws     = ws + 5 * SZ1 + (size_t)BT * NCH * NST;

  hipLaunchKernelGGL(k1_proj, dim3(BT * LSEQ / 64), dim3(128), 0, stream,
                     x, Wb, bb, Wc, bc, Wd, bd, Av, dA_ws, dB_ws, C_ws);
  hipLaunchKernelGGL(k2a_scan, dim3(BT * NCH), dim3(128), 0, stream,
                     x, dA_ws, dB_ws, C_ws, out, cumPC_ws, S_ws, P_ws);
  hipLaunchKernelGGL(k2b_combine, dim3(BT), dim3(128), 0, stream,
                     S_ws, P_ws, H_ws);
  hipLaunchKernelGGL(k3_corr, dim3(BT * NCH), dim3(256), 0, stream,
                     cumPC_ws, H_ws, out);
}